// Erode_86749749445290
// MI455X (gfx1250) — compile-verified
//
#include <hip/hip_runtime.h>
#include <cstdint>

// 5x5 grayscale erosion with zero padding (general se-bias form).
// Tile 64x16 outputs per 256-thread block; stage (64+4)x(16+4) fp32 tile into
// LDS via gfx1250 async global->LDS copies, then 25-tap min from LDS.

#define TW    64
#define TH    16
#define HALO  2
#define LW    (TW + 2 * HALO)      // 68 floats per LDS row
#define LH    (TH + 2 * HALO)      // 20 LDS rows
#define LDS_N (LW * LH)            // 1360 floats = 5440 B
#define BLOCK 256

__device__ __forceinline__ void async_copy_b32(uint32_t lds_byte_off,
                                               const float* sbase,
                                               uint32_t glob_byte_off) {
    // VDST = per-lane LDS byte offset, VADDR = per-lane global byte offset,
    // SADDR = uniform 64-bit base (SGPR pair). Tracked by ASYNCcnt.
    asm volatile("global_load_async_to_lds_b32 %0, %1, %2"
                 :
                 : "v"(lds_byte_off), "v"(glob_byte_off), "s"(sbase)
                 : "memory");
}

__global__ __launch_bounds__(BLOCK) void erode5x5_kernel(
    const float* __restrict__ x, const float* __restrict__ se,
    float* __restrict__ out, int H, int W)
{
    __shared__ float tile[LDS_N];

    const int plane   = blockIdx.z;
    const int tile_x  = blockIdx.x * TW;
    const int tile_y  = blockIdx.y * TH;
    const long plane_off = (long)plane * H * W;
    const float* xp = x + plane_off;          // uniform per block -> SGPR base

    // Low 32 bits of a generic pointer to LDS == workgroup LDS byte offset.
    const uint32_t lds0 = (uint32_t)(uintptr_t)(&tile[0]);

    // ---- Stage tile + halo: async copies for in-bounds, ds-store 0 for pad ----
    for (int li = threadIdx.x; li < LDS_N; li += BLOCK) {
        const int lr = li / LW;
        const int lc = li - lr * LW;
        const int gr = tile_y - HALO + lr;
        const int gc = tile_x - HALO + lc;
        const bool inb = (gr >= 0) && (gr < H) && (gc >= 0) && (gc < W);
        if (inb) {
            async_copy_b32(lds0 + (uint32_t)li * 4u, xp,
                           (uint32_t)(gr * W + gc) * 4u);
        } else {
            tile[li] = 0.0f;   // zero padding participates in the min
        }
    }

    // Wait for async LDS writes from this wave, then cross-wave barrier
    // (compiler inserts the dscnt wait for the zero-fill stores).
    asm volatile("s_wait_asynccnt 0x0" ::: "memory");
    __syncthreads();

    // ---- Per-tap constants from se: val = (active ? v : 0) - bias ----
    float bias[25];
    bool  act[25];
#pragma unroll
    for (int k = 0; k < 25; ++k) {
        const float b = se[k] - 1.0f;
        bias[k] = b;
        act[k]  = (b >= 0.0f);
    }

    // ---- Compute: 4 outputs per thread (rows ty, ty+4, ty+8, ty+12) ----
    const int tx = threadIdx.x & (TW - 1);   // 0..63
    const int ty = threadIdx.x >> 6;         // 0..3
    float* op = out + plane_off;

#pragma unroll
    for (int rr = 0; rr < TH; rr += 4) {
        const int r  = ty + rr;
        const int gr = tile_y + r;
        const int gc = tile_x + tx;
        if (gr < H && gc < W) {
            float acc = __builtin_inff();
#pragma unroll
            for (int dr = 0; dr < 5; ++dr) {
#pragma unroll
                for (int dc = 0; dc < 5; ++dc) {
                    const int   k = dr * 5 + dc;
                    const float v = tile[(r + dr) * LW + (tx + dc)];
                    const float val = (act[k] ? v : 0.0f) - bias[k];
                    acc = fminf(acc, val);
                }
            }
            op[gr * W + gc] = acc;
        }
    }
}

extern "C" void kernel_launch(void* const* d_in, const int* in_sizes, int n_in,
                              void* d_out, int out_size, void* d_ws, size_t ws_size,
                              hipStream_t stream) {
    const float* x  = (const float*)d_in[0];   // (8,3,512,512) fp32
    const float* se = (const float*)d_in[1];   // (5,5) fp32
    float* out = (float*)d_out;

    const int H = 512, W = 512;
    const int planes = in_sizes[0] / (H * W);  // 8*3 = 24

    dim3 grid((W + TW - 1) / TW, (H + TH - 1) / TH, planes); // 8 x 32 x 24
    erode5x5_kernel<<<grid, BLOCK, 0, stream>>>(x, se, out, H, W);
}